// GCN_36541581754764
// MI455X (gfx1250) — compile-verified
//
#include <hip/hip_runtime.h>

#define N_NODES  100000
#define N_EDGES  1600000
#define N_GRAPHS 128
#define FDIM     64
#define ODIM     32
#define WT_STR   66   // LDS row stride for transposed W: keeps 8B align, no bank conflicts

typedef __attribute__((ext_vector_type(2))) float v2f;
typedef __attribute__((ext_vector_type(8))) float v8f;

// ---------------------------------------------------------------- degrees
__global__ void k_degree(const int* __restrict__ src, const int* __restrict__ dst,
                         float* __restrict__ degs, float* __restrict__ degd) {
    int e = blockIdx.x * blockDim.x + threadIdx.x;
    if (e < N_EDGES) {
        atomicAdd(&degs[src[e]], 1.0f);
        atomicAdd(&degd[dst[e]], 1.0f);
    }
}

// deg -> deg^{-1/2} (0 if deg==0), in place
__global__ void k_norm(float* __restrict__ ns, float* __restrict__ nd) {
    int n = blockIdx.x * blockDim.x + threadIdx.x;
    if (n < N_NODES) {
        float a = ns[n]; ns[n] = (a > 0.0f) ? rsqrtf(a) : 0.0f;
        float b = nd[n]; nd[n] = (b > 0.0f) ? rsqrtf(b) : 0.0f;
    }
}

// hs = x * norm_src[node]  (float4 per thread)
__global__ void k_scale_x(const float* __restrict__ x, const float* __restrict__ ns,
                          float* __restrict__ hs) {
    int i = blockIdx.x * blockDim.x + threadIdx.x;      // N_NODES * 16
    if (i < N_NODES * (FDIM / 4)) {
        int n = i >> 4;
        float s = ns[n];
        float4 v = ((const float4*)x)[i];
        v.x *= s; v.y *= s; v.z *= s; v.w *= s;
        ((float4*)hs)[i] = v;
    }
}

// agg[dst[e]] += hs[src[e]]  -- one thread per (edge, float4 chunk)
__global__ void k_scatter(const float* __restrict__ hs, const int* __restrict__ src,
                          const int* __restrict__ dst, float* __restrict__ agg) {
    int idx = blockIdx.x * blockDim.x + threadIdx.x;    // N_EDGES * 16
    if (idx >= N_EDGES * (FDIM / 4)) return;
    int e  = idx >> 4;
    int f4 = (idx & 15) << 2;
    int s = src[e], d = dst[e];
    float4 v = *(const float4*)(hs + (size_t)s * FDIM + f4);
    float* o = agg + (size_t)d * FDIM + f4;
    atomicAdd(o + 0, v.x);
    atomicAdd(o + 1, v.y);
    atomicAdd(o + 2, v.z);
    atomicAdd(o + 3, v.w);
}

// out[m][:] = relu((agg[m][:]*norm_dst[m]) @ W + b) * (apply_src ? norm_src[m] : 1)
// One wave32 owns two 16-row tiles (32 rows), 8 v8f accumulators cover 32x64 output.
// K=64 via 16 steps of WMMA f32 16x16x4. W staged transposed in LDS so each lane's
// (k0,k0+1) B pair is one aligned ds_load_b64 directly into a WMMA source pair.
__global__ __launch_bounds__(256) void k_gemm(const float* __restrict__ agg,
                                              const float* __restrict__ W,
                                              const float* __restrict__ bias,
                                              const float* __restrict__ norm_dst,
                                              const float* __restrict__ norm_src,
                                              float* __restrict__ out,
                                              int apply_src) {
    __shared__ float Wt[FDIM * WT_STR];   // Wt[n*WT_STR + k] = W[k*FDIM + n]
    __shared__ float bl[FDIM];
    for (int i = threadIdx.x; i < FDIM * FDIM; i += 256) {
        int k = i >> 6, n = i & (FDIM - 1);
        Wt[n * WT_STR + k] = W[i];        // coalesced read, conflict-free write
    }
    if (threadIdx.x < FDIM) bl[threadIdx.x] = bias[threadIdx.x];
    __syncthreads();

    const int wave = threadIdx.x >> 5;    // wave32
    const int lane = threadIdx.x & 31;
    const int m0   = (blockIdx.x * 8 + wave) * 32;
    if (m0 >= N_NODES) return;

    const int mloc  = lane & 15;
    const int khalf = lane >> 4;          // 0: K pair {0,1}, 1: K pair {2,3}
    const int nloc  = lane & 15;

    const int rowA = m0 + mloc;
    const int rowB = m0 + 16 + mloc;
    const int rAc  = (rowA < N_NODES) ? rowA : (N_NODES - 1);
    const int rBc  = (rowB < N_NODES) ? rowB : (N_NODES - 1);
    const float ndA = (rowA < N_NODES) ? norm_dst[rAc] : 0.0f;  // zero kills OOB rows
    const float ndB = (rowB < N_NODES) ? norm_dst[rBc] : 0.0f;
    const float* __restrict__ aA = agg + (size_t)rAc * FDIM;
    const float* __restrict__ aB = agg + (size_t)rBc * FDIM;

    v8f acc[2][4];
    #pragma unroll
    for (int t = 0; t < 2; ++t)
        #pragma unroll
        for (int nt = 0; nt < 4; ++nt)
            acc[t][nt] = (v8f){};

    #pragma unroll
    for (int kb = 0; kb < 16; ++kb) {
        const int k0 = kb * 4 + khalf * 2;

        v2f a0; a0.x = aA[k0] * ndA; a0.y = aA[k0 + 1] * ndA;
        v2f a1; a1.x = aB[k0] * ndB; a1.y = aB[k0 + 1] * ndB;

        v2f b[4];
        #pragma unroll
        for (int nt = 0; nt < 4; ++nt)
            b[nt] = *(const v2f*)&Wt[(nt * 16 + nloc) * WT_STR + k0];  // aligned ds_load_b64

        #pragma unroll
        for (int nt = 0; nt < 4; ++nt) {
            acc[0][nt] = __builtin_amdgcn_wmma_f32_16x16x4_f32(
                false, a0, false, b[nt], (short)0, acc[0][nt], false, false);
            acc[1][nt] = __builtin_amdgcn_wmma_f32_16x16x4_f32(
                false, a1, false, b[nt], (short)0, acc[1][nt], false, false);
        }
    }

    // D layout: VGPR r holds M = r (lanes 0-15) / M = r+8 (lanes 16-31), N = lane&15
    const int mhi = khalf * 8;
    #pragma unroll
    for (int t = 0; t < 2; ++t) {
        #pragma unroll
        for (int r = 0; r < 8; ++r) {
            int m = m0 + t * 16 + mhi + r;
            if (m < N_NODES) {
                float sc = apply_src ? norm_src[m] : 1.0f;
                float* orow = out + (size_t)m * FDIM;
                #pragma unroll
                for (int nt = 0; nt < 4; ++nt) {
                    int n = nt * 16 + nloc;
                    orow[n] = fmaxf(acc[t][nt][r] + bl[n], 0.0f) * sc;
                }
            }
        }
    }
}

// per-graph sums + counts
__global__ void k_readout(const float* __restrict__ h, const int* __restrict__ gid,
                          float* __restrict__ gsum, float* __restrict__ gcnt) {
    int i = blockIdx.x * blockDim.x + threadIdx.x;      // N_NODES * 16
    if (i >= N_NODES * (FDIM / 4)) return;
    int n  = i >> 4;
    int f4 = (i & 15) << 2;
    int g  = gid[n];
    float4 v = *(const float4*)(h + (size_t)n * FDIM + f4);
    float* o = gsum + (size_t)g * FDIM + f4;
    atomicAdd(o + 0, v.x);
    atomicAdd(o + 1, v.y);
    atomicAdd(o + 2, v.z);
    atomicAdd(o + 3, v.w);
    if (f4 == 0) atomicAdd(&gcnt[g], 1.0f);
}

// out[g][o] = relu( (gsum[g]/max(cnt,1)) @ Wl + bl )
__global__ void k_final(const float* __restrict__ gsum, const float* __restrict__ gcnt,
                        const float* __restrict__ Wl, const float* __restrict__ bl,
                        float* __restrict__ out) {
    int i = blockIdx.x * blockDim.x + threadIdx.x;      // N_GRAPHS * ODIM
    if (i >= N_GRAPHS * ODIM) return;
    int g = i / ODIM, o = i % ODIM;
    float inv = 1.0f / fmaxf(gcnt[g], 1.0f);
    float s = bl[o];
    #pragma unroll 4
    for (int f = 0; f < FDIM; ++f)
        s = fmaf(gsum[g * FDIM + f] * inv, Wl[f * ODIM + o], s);
    out[i] = fmaxf(s, 0.0f);
}

extern "C" void kernel_launch(void* const* d_in, const int* in_sizes, int n_in,
                              void* d_out, int out_size, void* d_ws, size_t ws_size,
                              hipStream_t stream) {
    const float* x   = (const float*)d_in[0];
    const int*   src = (const int*)  d_in[1];
    const int*   dst = (const int*)  d_in[2];
    const int*   gid = (const int*)  d_in[3];
    const float* W1  = (const float*)d_in[4];
    const float* b1  = (const float*)d_in[5];
    const float* W2  = (const float*)d_in[6];
    const float* b2  = (const float*)d_in[7];
    const float* W3  = (const float*)d_in[8];
    const float* b3  = (const float*)d_in[9];
    const float* Wl  = (const float*)d_in[10];
    const float* blb = (const float*)d_in[11];

    // workspace carve-up (256B aligned)
    char* ws = (char*)d_ws;
    size_t off = 0;
    auto alloc = [&](size_t bytes) { char* p = ws + off; off = (off + bytes + 255) & ~(size_t)255; return p; };
    float* ns   = (float*)alloc((size_t)N_NODES * sizeof(float));
    float* nd   = (float*)alloc((size_t)N_NODES * sizeof(float));
    float* hs   = (float*)alloc((size_t)N_NODES * FDIM * sizeof(float));
    float* agg  = (float*)alloc((size_t)N_NODES * FDIM * sizeof(float));
    float* gsum = (float*)alloc((size_t)N_GRAPHS * FDIM * sizeof(float));
    float* gcnt = (float*)alloc((size_t)N_GRAPHS * sizeof(float));
    (void)ws_size; (void)n_in; (void)in_sizes; (void)out_size;

    const int T = 256;
    const int gEdges = (N_EDGES + T - 1) / T;
    const int gNodes = (N_NODES + T - 1) / T;
    const int gElem4 = (N_NODES * (FDIM / 4) + T - 1) / T;
    const int gScat  = (N_EDGES * (FDIM / 4) + T - 1) / T;
    const int gGemm  = (N_NODES + 255) / 256;           // 8 waves x 32 rows per block

    // degrees -> norms
    hipMemsetAsync(ns, 0, (size_t)N_NODES * sizeof(float), stream);
    hipMemsetAsync(nd, 0, (size_t)N_NODES * sizeof(float), stream);
    k_degree<<<gEdges, T, 0, stream>>>(src, dst, ns, nd);
    k_norm<<<gNodes, T, 0, stream>>>(ns, nd);

    // hs = x * norm_src
    k_scale_x<<<gElem4, T, 0, stream>>>(x, ns, hs);

    const float* Ws[3] = {W1, W2, W3};
    const float* bs[3] = {b1, b2, b3};
    for (int l = 0; l < 3; ++l) {
        hipMemsetAsync(agg, 0, (size_t)N_NODES * FDIM * sizeof(float), stream);
        k_scatter<<<gScat, T, 0, stream>>>(hs, src, dst, agg);
        k_gemm<<<gGemm, T, 0, stream>>>(agg, Ws[l], bs[l], nd, ns, hs, (l < 2) ? 1 : 0);
    }

    // readout + final dense layer
    hipMemsetAsync(gsum, 0, (size_t)N_GRAPHS * FDIM * sizeof(float), stream);
    hipMemsetAsync(gcnt, 0, (size_t)N_GRAPHS * sizeof(float), stream);
    k_readout<<<gElem4, T, 0, stream>>>(hs, gid, gsum, gcnt);
    k_final<<<(N_GRAPHS * ODIM + T - 1) / T, T, 0, stream>>>(gsum, gcnt, Wl, blb, (float*)d_out);
}